// LightGCN_49787260895316
// MI455X (gfx1250) — compile-verified
//
#include <hip/hip_runtime.h>

typedef float v2f __attribute__((ext_vector_type(2)));
typedef float v8f __attribute__((ext_vector_type(8)));

#define TPB 256
#define EMB 64
#define NLAYERS 3

// ---------------- init: x = acc = concat(user_emb, item_emb); deg = 0 ----------------
__global__ void lg_init(const float* __restrict__ ue, const float* __restrict__ ie,
                        float* __restrict__ x, float* __restrict__ acc,
                        float* __restrict__ deg, int nu_elems, int total_elems, int nnodes) {
  int t = blockIdx.x * blockDim.x + threadIdx.x;
  if (t < total_elems) {
    float v = (t < nu_elems) ? ue[t] : ie[t - nu_elems];
    x[t] = v;
    acc[t] = v;
  }
  if (t < nnodes) deg[t] = 0.0f;
}

// ---------------- degree accumulation: deg = segment_sum(ones, edge_row) ----------------
__global__ void lg_degree(const int* __restrict__ er, float* __restrict__ deg, int E) {
  int t = blockIdx.x * blockDim.x + threadIdx.x;
  if (t < E) unsafeAtomicAdd(&deg[er[t]], 1.0f);
}

// ---------------- deg -> d^{-1/2} in place ----------------
__global__ void lg_rsqrt(float* __restrict__ deg, int n) {
  int t = blockIdx.x * blockDim.x + threadIdx.x;
  if (t < n) {
    float d = deg[t];
    deg[t] = (d > 0.0f) ? rsqrtf(d) : 0.0f;
  }
}

// ---------------- zero buffer ----------------
__global__ void lg_zero(float* __restrict__ y, int total_elems) {
  int t = blockIdx.x * blockDim.x + threadIdx.x;
  if (t < total_elems) y[t] = 0.0f;
}

// ---------------- SpMM: y[row] += dinv[row]*dinv[col] * x[col] ----------------
// 16 threads per edge, 4 consecutive floats per thread (b128 gather, 4 f32 atomics).
__global__ void lg_spmm(const int* __restrict__ er, const int* __restrict__ ec,
                        const float* __restrict__ dinv,
                        const float* __restrict__ xin, float* __restrict__ yout, int E) {
  unsigned t = blockIdx.x * blockDim.x + threadIdx.x;
  unsigned e = t >> 4;
  if (e >= (unsigned)E) return;
  int c4 = (int)(t & 15u) * 4;
  int r = er[e];
  int c = ec[e];
  float v = dinv[r] * dinv[c];
  const float4 xv = *reinterpret_cast<const float4*>(xin + (size_t)c * EMB + c4);
  float* yp = yout + (size_t)r * EMB + c4;
  unsafeAtomicAdd(yp + 0, v * xv.x);
  unsafeAtomicAdd(yp + 1, v * xv.y);
  unsafeAtomicAdd(yp + 2, v * xv.z);
  unsafeAtomicAdd(yp + 3, v * xv.w);
}

// ---------------- acc += y ----------------
__global__ void lg_accum(float* __restrict__ acc, const float* __restrict__ y, int total_elems) {
  int t = blockIdx.x * blockDim.x + threadIdx.x;
  if (t < total_elems) acc[t] += y[t];
}

// ---------------- scoring via V_WMMA_F32_16X16X4_F32 ----------------
// One wave handles 16 (user,item) pairs: D = U(16x64) x I^T(64x16) accumulated
// over 16 K=4 steps; scores are the diagonal of D. Full f32 precision.
__global__ void lg_score(const float* __restrict__ acc, const int* __restrict__ uidx,
                         const int* __restrict__ iidx, float* __restrict__ out,
                         int NU, int B) {
  int gid = blockIdx.x * blockDim.x + threadIdx.x;
  int wave = gid >> 5;
  int lane = gid & 31;
  int base = wave * 16;
  if (base >= B) return;  // wave-uniform exit: EXEC stays all-ones for WMMA

  int m = lane & 15;
  int u = uidx[base + m];
  int it = iidx[base + m] + NU;
  const float* ar = acc + (size_t)u * EMB;   // A[m][k] = ue row m
  const float* br = acc + (size_t)it * EMB;  // B[k][n] = ie row n (transposed feed)
  int kb = (lane < 16) ? 0 : 2;              // K-half split across lane halves (ISA layout)

  v8f c = {0.f, 0.f, 0.f, 0.f, 0.f, 0.f, 0.f, 0.f};
#pragma unroll
  for (int kk = 0; kk < 16; ++kk) {
    v2f a, b;
    a[0] = ar[kk * 4 + kb];
    a[1] = ar[kk * 4 + kb + 1];
    b[0] = br[kk * 4 + kb];
    b[1] = br[kk * 4 + kb + 1];
    c = __builtin_amdgcn_wmma_f32_16x16x4_f32(false, a, false, b, (short)0, c, false, false);
  }

  // Diagonal extraction from 16x16 f32 C/D layout:
  //  lanes 0-15:  VGPR j holds (M=j,   N=lane)    -> diag for lanes 0-7,  j=lane
  //  lanes 16-31: VGPR j holds (M=j+8, N=lane-16) -> diag for lanes 24-31, j=lane-24
  bool valid = (lane < 8) || (lane >= 24);
  int idx = (lane < 8) ? lane : (lane - 24);
  float s = c[0];
  if (idx == 1) s = c[1];
  if (idx == 2) s = c[2];
  if (idx == 3) s = c[3];
  if (idx == 4) s = c[4];
  if (idx == 5) s = c[5];
  if (idx == 6) s = c[6];
  if (idx == 7) s = c[7];
  if (valid) {
    int mm = (lane < 8) ? lane : (lane - 16);
    // final = acc/(L+1); score = dot(final_u, final_i) = dot(acc_u, acc_i)/16
    out[base + mm] = s * 0.0625f;
  }
}

extern "C" void kernel_launch(void* const* d_in, const int* in_sizes, int n_in,
                              void* d_out, int out_size, void* d_ws, size_t ws_size,
                              hipStream_t stream) {
  const float* user_emb = (const float*)d_in[0];
  const float* item_emb = (const float*)d_in[1];
  const int* edge_row   = (const int*)d_in[2];
  const int* edge_col   = (const int*)d_in[3];
  const int* user_idx   = (const int*)d_in[4];
  const int* item_idx   = (const int*)d_in[5];
  float* out = (float*)d_out;

  const int NU = in_sizes[0] / EMB;
  const int NI = in_sizes[1] / EMB;
  const int NN = NU + NI;
  const int E  = in_sizes[2];
  const int B  = in_sizes[4];

  const size_t node_elems = (size_t)NN * EMB;
  float* ws  = (float*)d_ws;
  float* x   = ws;
  float* y   = x + node_elems;
  float* acc = y + node_elems;
  float* deg = acc + node_elems;  // becomes d^{-1/2}

  const int total = (int)node_elems;
  const int gb_nodeE = (total + TPB - 1) / TPB;
  const int gb_nodes = (NN + TPB - 1) / TPB;
  const int gb_edges = (E + TPB - 1) / TPB;
  const int gb_spmm  = (E * 16 + TPB - 1) / TPB;

  // init x = acc = concat(embeddings); deg = 0
  lg_init<<<gb_nodeE, TPB, 0, stream>>>(user_emb, item_emb, x, acc, deg,
                                        NU * EMB, total, NN);
  // degrees and d^{-1/2}
  lg_degree<<<gb_edges, TPB, 0, stream>>>(edge_row, deg, E);
  lg_rsqrt<<<gb_nodes, TPB, 0, stream>>>(deg, NN);

  // 3 propagation layers with pointer ping-pong
  float* cur = x;
  float* nxt = y;
  for (int l = 0; l < NLAYERS; ++l) {
    lg_zero<<<gb_nodeE, TPB, 0, stream>>>(nxt, total);
    lg_spmm<<<gb_spmm, TPB, 0, stream>>>(edge_row, edge_col, deg, cur, nxt, E);
    lg_accum<<<gb_nodeE, TPB, 0, stream>>>(acc, nxt, total);
    float* tmp = cur; cur = nxt; nxt = tmp;
  }

  // scoring: 16 pairs per wave, ceil(B/16) waves
  const int waves = (B + 15) / 16;
  const int sthreads = waves * 32;
  const int gb_score = (sthreads + TPB - 1) / TPB;
  lg_score<<<gb_score, TPB, 0, stream>>>(acc, user_idx, item_idx, out, NU, B);
}